// HippocampalFormation_26379689132267
// MI455X (gfx1250) — compile-verified
//
#include <hip/hip_runtime.h>
#include <math.h>

typedef __attribute__((ext_vector_type(2))) float v2f;
typedef __attribute__((ext_vector_type(4))) float v4f;
typedef __attribute__((ext_vector_type(8))) float v8f;

#define BLK   256
#define WAVES 8      // 256 threads / wave32
#define NB2   120    // stage-1 top-k blocks

// ---------------------------------------------------------------------------
// Kernel 0: q_norm = q / max(||q||, eps)   (one block)
// ---------------------------------------------------------------------------
__global__ void qnorm_kernel(const float* __restrict__ q, float* __restrict__ qn, int D) {
  __shared__ float red[BLK];
  int tid = threadIdx.x;
  float ss = 0.f;
  for (int i = tid; i < D; i += BLK) { float v = q[i]; ss = fmaf(v, v, ss); }
  red[tid] = ss; __syncthreads();
  for (int off = BLK / 2; off > 0; off >>= 1) {
    if (tid < off) red[tid] += red[tid + off];
    __syncthreads();
  }
  float inv = 1.0f / fmaxf(sqrtf(red[0]), 1e-12f);
  for (int i = tid; i < D; i += BLK) qn[i] = q[i] * inv;
}

// ---------------------------------------------------------------------------
// Kernel 1: WMMA-accumulated GEMV + per-row scoring.
// Each wave owns a 16-row tile. Per K-step of 8, one nontemporal
// global_load_b128 of A and one ds_load_b128 of q feed TWO
// V_WMMA_F32_16X16X4_F32 ops (K-slot permutation is legal for a dot
// product as long as A and B agree). Row sum-of-squares accumulates on
// the VALU (co-executes with XDL); K-halves combined via shfl_xor(16).
// Non-column-0 lanes read B from a zeroed LDS region -> no per-step masking.
// ---------------------------------------------------------------------------
__global__ void score_kernel(const float* __restrict__ mf,
                             const float* __restrict__ mloc,
                             const float* __restrict__ meta,
                             const float* __restrict__ qn,
                             const float* __restrict__ loc,
                             float* __restrict__ comb, int M) {
  const int D = 768;
  __shared__ float qs[2 * 768];        // [0..767] = q_norm, [768..1535] = zeros
  __shared__ float simbuf[WAVES * 16];

  int tid = threadIdx.x;
  for (int i = tid; i < D; i += BLK) { qs[i] = qn[i]; qs[D + i] = 0.f; }
  __syncthreads();

  int wave   = tid >> 5;
  int lane   = tid & 31;
  int laneHi = lane >> 4;   // which K-half of the slot space this lane feeds
  int laneM  = lane & 15;   // row within tile

  int numTiles = (M + 15) >> 4;
  int tile  = blockIdx.x * WAVES + wave;
  bool valid = tile < numTiles;
  int t = valid ? tile : 0;            // keep EXEC all-ones for WMMA
  int rowBase = t << 4;

  long arow = rowBase + laneM;
  if (arow >= M) arow = M - 1;         // clamp tail reads
  const bool bsel = (laneM == 0);      // lanes 0 and 16 carry q in B column 0
  const float* aptr = mf + (size_t)arow * D + laneHi * 4;
  const float* qptr = qs + (bsel ? 0 : D) + laneHi * 4;

  v8f  c  = {};                        // 16x16 f32 accumulator (column 0 useful)
  float ss = 0.f;

  #pragma unroll 4
  for (int k = 0; k < D; k += 8) {
    v4f a  = __builtin_nontemporal_load(reinterpret_cast<const v4f*>(aptr + k));
    v4f qv = *reinterpret_cast<const v4f*>(qptr + k);   // ds_load_b128
    v2f a0; a0.x = a.x; a0.y = a.y;
    v2f a1; a1.x = a.z; a1.y = a.w;
    v2f b0; b0.x = qv.x; b0.y = qv.y;
    v2f b1; b1.x = qv.z; b1.y = qv.w;
    c = __builtin_amdgcn_wmma_f32_16x16x4_f32(false, a0, false, b0,
                                              (short)0, c, false, false);
    c = __builtin_amdgcn_wmma_f32_16x16x4_f32(false, a1, false, b1,
                                              (short)0, c, false, false);
    ss = fmaf(a.x, a.x, ss);
    ss = fmaf(a.y, a.y, ss);
    ss = fmaf(a.z, a.z, ss);
    ss = fmaf(a.w, a.w, ss);
  }

  // combine K-halves of the sum-of-squares: lane L and L+16 both get row L%16
  ss += __shfl_xor(ss, 16, 32);

  // D column 0: lane0 holds rows 0..7 in c[0..7]; lane16 holds rows 8..15
  float* sb = &simbuf[wave * 16];
  if (laneM == 0) {
    #pragma unroll
    for (int j = 0; j < 8; j++) sb[laneHi * 8 + j] = c[j];
  }
  __asm volatile("s_wait_dscnt 0" ::: "memory");

  if (valid && lane < 16) {
    int row = rowBase + lane;
    if (row < M) {
      float dot = sb[lane];
      float sim = dot / fmaxf(sqrtf(ss), 1e-12f);
      float dx = mloc[row * 2 + 0] - loc[0];
      float dy = mloc[row * 2 + 1] - loc[1];
      float spatial  = 1.0f / (1.0f + sqrtf(fmaf(dx, dx, dy * dy)));
      float strength = meta[row * 4 + 0];
      float ts       = meta[row * 4 + 1];
      float temporal = __expf((ts - 3600.0f) * (1.0f / 3600.0f));
      comb[row] = (0.5f * sim + 0.3f * spatial + 0.2f * temporal) * strength;
    }
  }
}

// ---------------------------------------------------------------------------
// Top-K: stage 1 (per-block top-K of a grid-strided chunk)
// ---------------------------------------------------------------------------
template <int KK>
__global__ void topk_partial(const float* __restrict__ comb, int M,
                             float* __restrict__ candS, int* __restrict__ candI) {
  __shared__ float sv[BLK]; __shared__ int si[BLK]; __shared__ int st[BLK];
  int tid = threadIdx.x;
  float s[KK]; int id[KK];
  #pragma unroll
  for (int j = 0; j < KK; j++) { s[j] = -INFINITY; id[j] = -1; }

  for (int i = blockIdx.x * BLK + tid; i < M; i += gridDim.x * BLK) {
    float v = comb[i]; int ix = i;
    if (v > s[KK - 1]) {
      #pragma unroll
      for (int j = 0; j < KK; j++) {          // static-index bubble insert
        if (v > s[j]) {
          float tS = s[j]; int tI = id[j];
          s[j] = v; id[j] = ix;
          v = tS; ix = tI;
        }
      }
    }
  }

  for (int r = 0; r < KK; r++) {
    sv[tid] = s[0]; si[tid] = id[0]; st[tid] = tid;
    __syncthreads();
    for (int off = BLK / 2; off > 0; off >>= 1) {
      if (tid < off && sv[tid + off] > sv[tid]) {
        sv[tid] = sv[tid + off]; si[tid] = si[tid + off]; st[tid] = st[tid + off];
      }
      __syncthreads();
    }
    if (tid == 0) { candS[blockIdx.x * KK + r] = sv[0]; candI[blockIdx.x * KK + r] = si[0]; }
    int w = st[0];
    __syncthreads();
    if (tid == w) {
      #pragma unroll
      for (int j = 0; j < KK - 1; j++) { s[j] = s[j + 1]; id[j] = id[j + 1]; }
      s[KK - 1] = -INFINITY; id[KK - 1] = -1;
    }
  }
}

// ---------------------------------------------------------------------------
// Top-K: stage 2 (single block merges all candidates, writes output)
// out[0..K-1] = scores, out[K..2K-1] = indices (as float)
// ---------------------------------------------------------------------------
template <int KK>
__global__ void topk_final(const float* __restrict__ candS, const int* __restrict__ candI,
                           int N, float* __restrict__ out) {
  __shared__ float sv[BLK]; __shared__ int si[BLK]; __shared__ int st[BLK];
  int tid = threadIdx.x;
  float s[KK]; int id[KK];
  #pragma unroll
  for (int j = 0; j < KK; j++) { s[j] = -INFINITY; id[j] = -1; }

  for (int i = tid; i < N; i += BLK) {
    float v = candS[i]; int ix = candI[i];
    if (v > s[KK - 1]) {
      #pragma unroll
      for (int j = 0; j < KK; j++) {
        if (v > s[j]) {
          float tS = s[j]; int tI = id[j];
          s[j] = v; id[j] = ix;
          v = tS; ix = tI;
        }
      }
    }
  }

  for (int r = 0; r < KK; r++) {
    sv[tid] = s[0]; si[tid] = id[0]; st[tid] = tid;
    __syncthreads();
    for (int off = BLK / 2; off > 0; off >>= 1) {
      if (tid < off && sv[tid + off] > sv[tid]) {
        sv[tid] = sv[tid + off]; si[tid] = si[tid + off]; st[tid] = st[tid + off];
      }
      __syncthreads();
    }
    if (tid == 0) { out[r] = sv[0]; out[KK + r] = (float)si[0]; }
    int w = st[0];
    __syncthreads();
    if (tid == w) {
      #pragma unroll
      for (int j = 0; j < KK - 1; j++) { s[j] = s[j + 1]; id[j] = id[j + 1]; }
      s[KK - 1] = -INFINITY; id[KK - 1] = -1;
    }
  }
}

// ---------------------------------------------------------------------------
extern "C" void kernel_launch(void* const* d_in, const int* in_sizes, int n_in,
                              void* d_out, int out_size, void* d_ws, size_t ws_size,
                              hipStream_t stream) {
  const float* q    = (const float*)d_in[0];   // (768,)
  const float* loc  = (const float*)d_in[1];   // (2,)
  const float* mf   = (const float*)d_in[2];   // (M,768)
  const float* mloc = (const float*)d_in[3];   // (M,2)
  const float* meta = (const float*)d_in[4];   // (M,4)
  float* out = (float*)d_out;

  const int D = 768;
  const int M = in_sizes[2] / D;
  const int K = out_size / 2;                  // tuple (scores, indices)

  // workspace layout
  float* wsf   = (float*)d_ws;
  float* qn    = wsf;                          // 768 (padded to 1024)
  float* comb  = wsf + 1024;                   // M
  float* candS = comb + M;                     // NB2*8 max
  int*   candI = (int*)(candS + NB2 * 8);      // NB2*8 max

  qnorm_kernel<<<1, BLK, 0, stream>>>(q, qn, D);

  int numTiles = (M + 15) / 16;
  int blocks1  = (numTiles + WAVES - 1) / WAVES;
  score_kernel<<<blocks1, BLK, 0, stream>>>(mf, mloc, meta, qn, loc, comb, M);

  #define LAUNCH_TOPK(KK)                                                     \
    do {                                                                      \
      topk_partial<KK><<<NB2, BLK, 0, stream>>>(comb, M, candS, candI);       \
      topk_final<KK><<<1, BLK, 0, stream>>>(candS, candI, NB2 * (KK), out);   \
    } while (0)

  switch (K) {
    case 1: LAUNCH_TOPK(1); break;
    case 2: LAUNCH_TOPK(2); break;
    case 3: LAUNCH_TOPK(3); break;
    case 4: LAUNCH_TOPK(4); break;
    case 6: LAUNCH_TOPK(6); break;
    case 7: LAUNCH_TOPK(7); break;
    case 8: LAUNCH_TOPK(8); break;
    default: LAUNCH_TOPK(5); break;            // reference k = 5
  }
  #undef LAUNCH_TOPK
}